// VolumeRenderer_84602265796720
// MI455X (gfx1250) — compile-verified
//
#include <hip/hip_runtime.h>
#include <cstdint>

// ---- problem constants (from reference) ----
#define TREE_N      2
#define DEPTH       2
#define DATA_DIM    28
#define BASIS_DIM   9
#define MAX_STEPS   16
#define STEP_SIZE   0.001f
#define BG          1.0f

#define N_NODES     9                       // 1 + 2^3
#define N_CELLS     (N_NODES * 8)           // 72
#define DATA_FLOATS (N_CELLS * DATA_DIM)    // 2016 floats = 8064 B (16B-aligned chunks: 112B/cell)
#define DATA_B128   (DATA_FLOATS * 4 / 16)  // 504 16-byte chunks
#define CHILD_B128  (N_CELLS * 4 / 16)      // 18  16-byte chunks

#define SH_C0  0.28209479177387814f
#define SH_C1  0.4886025119029199f
#define SH_C2a 1.0925484305920792f
#define SH_C2b (-1.0925484305920792f)
#define SH_C2c 0.31539156525252005f
#define SH_C2d (-1.0925484305920792f)
#define SH_C2e 0.5462742152960396f

__global__ __launch_bounds__(256)
void volrender_kernel(const float* __restrict__ origins,
                      const float* __restrict__ dirs,
                      const float* __restrict__ viewdirs,
                      const float* __restrict__ data,
                      const int*   __restrict__ child,
                      const float* __restrict__ offset,
                      const float* __restrict__ invradius,
                      float* __restrict__ out,
                      int nrays)
{
    __shared__ __align__(16) float s_data[DATA_FLOATS];
    __shared__ __align__(16) int   s_child[N_CELLS];

    const int tid = threadIdx.x;

    // ---- CDNA5 async global->LDS copy of the octree (ASYNCcnt path) ----
    {
        const uint64_t gdata  = (uint64_t)(uintptr_t)data;
        const uint64_t gchild = (uint64_t)(uintptr_t)child;
        const uint32_t ldata  = (uint32_t)(uintptr_t)s_data;   // low 32 bits of generic = LDS offset
        const uint32_t lchild = (uint32_t)(uintptr_t)s_child;
        for (int i = tid; i < DATA_B128; i += 256) {
            uint32_t dst = ldata + (uint32_t)i * 16u;
            uint64_t src = gdata + (uint64_t)i * 16u;
            asm volatile("global_load_async_to_lds_b128 %0, %1, off"
                         :: "v"(dst), "v"(src) : "memory");
        }
        if (tid < CHILD_B128) {
            uint32_t dst = lchild + (uint32_t)tid * 16u;
            uint64_t src = gchild + (uint64_t)tid * 16u;
            asm volatile("global_load_async_to_lds_b128 %0, %1, off"
                         :: "v"(dst), "v"(src) : "memory");
        }
        asm volatile("s_wait_asynccnt 0" ::: "memory");
    }
    __syncthreads();

    const int ray = blockIdx.x * 256 + tid;
    if (ray >= nrays) return;

    // ---- per-ray setup ----
    const float ox = origins[ray*3+0], oy = origins[ray*3+1], oz = origins[ray*3+2];
    const float dx = dirs[ray*3+0],    dy = dirs[ray*3+1],    dz = dirs[ray*3+2];
    const float vx = viewdirs[ray*3+0],vy = viewdirs[ray*3+1],vz = viewdirs[ray*3+2];
    const float of0 = offset[0], of1 = offset[1], of2 = offset[2];
    const float invr = invradius[0];

    const float cx = of0 + ox * invr;
    const float cy = of1 + oy * invr;
    const float cz = of2 + oz * invr;

    const float rnorm = rsqrtf(dx*dx + dy*dy + dz*dz);
    const float dnx = dx * rnorm, dny = dy * rnorm, dnz = dz * rnorm;

    const float iv0 = __builtin_amdgcn_rcpf(dnx + 1e-9f);
    const float iv1 = __builtin_amdgcn_rcpf(dny + 1e-9f);
    const float iv2 = __builtin_amdgcn_rcpf(dnz + 1e-9f);

    // dda_unit(origins_t, invdirs)
    float t1x = -cx * iv0, t2x = t1x + iv0;
    float t1y = -cy * iv1, t2y = t1y + iv1;
    float t1z = -cz * iv2, t2z = t1z + iv2;
    float t    = fmaxf(fmaxf(fmaxf(fminf(t1x,t2x), fminf(t1y,t2y)), fminf(t1z,t2z)), 0.0f);
    const float tmax = fminf(fminf(fminf(fmaxf(t1x,t2x), fmaxf(t1y,t2y)), fmaxf(t1z,t2z)), 1e9f);

    // SH9 basis of viewdir
    float sh[9];
    sh[0] = SH_C0;
    sh[1] = -SH_C1 * vy;
    sh[2] =  SH_C1 * vz;
    sh[3] = -SH_C1 * vx;
    sh[4] = SH_C2a * vx * vy;
    sh[5] = SH_C2b * vy * vz;
    sh[6] = SH_C2c * (2.0f*vz*vz - vx*vx - vy*vy);
    sh[7] = SH_C2d * vx * vz;
    sh[8] = SH_C2e * (vx*vx - vy*vy);

    const float delta_scale = __builtin_amdgcn_rcpf(invr);

    float light = 1.0f;
    float o0 = 0.0f, o1 = 0.0f, o2 = 0.0f;

    for (int step = 0; step < MAX_STEPS; ++step) {
        const bool active = t < tmax;

        const float px = fmaf(t, dnx, cx);
        const float py = fmaf(t, dny, cy);
        const float pz = fmaf(t, dnz, cz);
        float qx = fminf(fmaxf(px, 0.0f), 1.0f - 1e-6f);
        float qy = fminf(fmaxf(py, 0.0f), 1.0f - 1e-6f);
        float qz = fminf(fmaxf(pz, 0.0f), 1.0f - 1e-6f);

        // ---- tree query (DEPTH=2, unrolled) ----
        int   node = 0, vbase = 0;
        float crx = 0.0f, cry = 0.0f, crz = 0.0f;
        float len = 1.0f;
        bool  done = false;
        float curx = qx, cury = qy, curz = qz;
        #pragma unroll
        for (int lev = 0; lev < DEPTH; ++lev) {
            int ix = (int)floorf(curx * 2.0f); ix = ix < 0 ? 0 : (ix > 1 ? 1 : ix);
            int iy = (int)floorf(cury * 2.0f); iy = iy < 0 ? 0 : (iy > 1 ? 1 : iy);
            int iz = (int)floorf(curz * 2.0f); iz = iz < 0 ? 0 : (iz > 1 ? 1 : iz);
            const int cell = node * 8 + ((ix * 2 + iy) * 2 + iz);
            const int ch   = s_child[cell];
            const float nl = len * 0.5f;
            if (!done) {
                vbase = cell * DATA_DIM;
                crx = fmaf((float)ix, nl, crx);
                cry = fmaf((float)iy, nl, cry);
                crz = fmaf((float)iz, nl, crz);
                len = nl;
                curx = curx * 2.0f - (float)ix;
                cury = cury * 2.0f - (float)iy;
                curz = curz * 2.0f - (float)iz;
                node += ch;
            }
            done = done || (ch == 0);
        }

        // pos_local = (pos - corner) / length   (uses UNCLAMPED pos, like reference)
        const float rlen = __builtin_amdgcn_rcpf(len);   // exact: len is a power of two
        const float plx = (px - crx) * rlen;
        const float ply = (py - cry) * rlen;
        const float plz = (pz - crz) * rlen;

        float a1x = -plx * iv0, a2x = a1x + iv0;
        float a1y = -ply * iv1, a2y = a1y + iv1;
        float a1z = -plz * iv2, a2z = a1z + iv2;
        float smin = fmaxf(fmaxf(fmaxf(fminf(a1x,a2x), fminf(a1y,a2y)), fminf(a1z,a2z)), 0.0f);
        float smax = fminf(fminf(fminf(fmaxf(a1x,a2x), fmaxf(a1y,a2y)), fmaxf(a1z,a2z)), 1e9f);
        const float dt = fmaf(smax - smin, len, STEP_SIZE);

        // ---- fetch the 28-float leaf payload from LDS as 7x ds_load_b128 ----
        __align__(16) float vv[DATA_DIM];
        const float4* cellp = (const float4*)(&s_data[vbase]);
        #pragma unroll
        for (int j = 0; j < 7; ++j) ((float4*)vv)[j] = cellp[j];

        const float sigma = fmaxf(vv[27], 0.0f);
        const float att   = __expf(-dt * sigma * delta_scale);
        const float weight = light * (1.0f - att);

        float rgb[3];
        #pragma unroll
        for (int c = 0; c < 3; ++c) {
            float sacc = 0.0f;
            #pragma unroll
            for (int k = 0; k < BASIS_DIM; ++k)
                sacc = fmaf(sh[k], vv[c * BASIS_DIM + k], sacc);
            rgb[c] = __builtin_amdgcn_rcpf(1.0f + __expf(-sacc));
        }

        const float w = active ? weight : 0.0f;
        o0 = fmaf(w, rgb[0], o0);
        o1 = fmaf(w, rgb[1], o1);
        o2 = fmaf(w, rgb[2], o2);
        if (active) { light *= att; t += dt; }
    }

    out[ray*3+0] = fmaf(light, BG, o0);
    out[ray*3+1] = fmaf(light, BG, o1);
    out[ray*3+2] = fmaf(light, BG, o2);
}

extern "C" void kernel_launch(void* const* d_in, const int* in_sizes, int n_in,
                              void* d_out, int out_size, void* d_ws, size_t ws_size,
                              hipStream_t stream) {
    (void)n_in; (void)out_size; (void)d_ws; (void)ws_size;
    const float* origins   = (const float*)d_in[0];
    const float* dirs      = (const float*)d_in[1];
    const float* viewdirs  = (const float*)d_in[2];
    const float* data      = (const float*)d_in[3];
    const int*   child     = (const int*)  d_in[4];
    const float* offset    = (const float*)d_in[5];
    const float* invradius = (const float*)d_in[6];
    float*       out       = (float*)d_out;

    const int nrays = in_sizes[0] / 3;
    const int blocks = (nrays + 255) / 256;
    volrender_kernel<<<blocks, 256, 0, stream>>>(origins, dirs, viewdirs, data, child,
                                                 offset, invradius, out, nrays);
}